// GNNModel_4037269258457
// MI455X (gfx1250) — compile-verified
//
#include <hip/hip_runtime.h>
#include <math.h>

typedef __attribute__((ext_vector_type(2))) float v2f;
typedef __attribute__((ext_vector_type(8))) float v8f;

#define DIN 128
#define DH  128

// ---------------------------------------------------------------- degree / inv
__global__ __launch_bounds__(256) void deg_kernel(const int* __restrict__ dst,
                                                  float* __restrict__ deg, int nE) {
    int e = blockIdx.x * 256 + threadIdx.x;
    if (e < nE) atomicAdd(&deg[dst[e]], 1.0f);
}

__global__ __launch_bounds__(256) void inv_kernel(float* __restrict__ deg, int nN) {
    int n = blockIdx.x * 256 + threadIdx.x;
    if (n < nN) deg[n] = 1.0f / fmaxf(deg[n], 1.0f);
}

// --------------------------------------------------------- edge scatter (mean numerator)
// One wave (32 lanes) per edge; each lane handles 4 consecutive features.
__global__ __launch_bounds__(256) void scatter_kernel(const float* __restrict__ xin,
                                                      const int* __restrict__ src,
                                                      const int* __restrict__ dst,
                                                      float* __restrict__ agg, int nE) {
    const int lane = threadIdx.x & 31;
    const int e = blockIdx.x * 8 + (threadIdx.x >> 5);
    if (e >= nE) return;
    const int s = src[e];
    const int d = dst[e];
    const float4 v = *(const float4*)(xin + (size_t)s * DH + lane * 4);
    float* ap = agg + (size_t)d * DH + lane * 4;
    atomicAdd(ap + 0, v.x);
    atomicAdd(ap + 1, v.y);
    atomicAdd(ap + 2, v.z);
    atomicAdd(ap + 3, v.w);
}

// --------------------------------------------------------- fused SAGE GEMM (WMMA f32)
// out[n,:] = act( (aggsum[n,:] * inv[n]) @ Wl + b + xin[n,:] @ Wr )
// One wave computes one 16x16 output tile; 8 waves/block cover one 16-row strip.
__global__ __launch_bounds__(256) void sage_gemm_kernel(const float* __restrict__ aggsum,
                                                        const float* __restrict__ xin,
                                                        const float* __restrict__ Wl,
                                                        const float* __restrict__ bias,
                                                        const float* __restrict__ Wr,
                                                        const float* __restrict__ inv,
                                                        float* __restrict__ out,
                                                        int nN) {
    const int lane = threadIdx.x & 31;
    const int wave = threadIdx.x >> 5;
    const int tile_id = blockIdx.x * 8 + wave;
    const int tile_m = tile_id >> 3;     // node strip
    const int tile_n = tile_id & 7;      // 16-col strip of 128
    const int row_base = tile_m * 16;
    if (row_base >= nN) return;          // wave-uniform (nN % 16 == 0 in practice)

    const int lm  = lane & 15;
    const int hi  = lane >> 4;           // 0: K0/K1 ; 1: K2/K3
    const int arow = row_base + lm;      // A-matrix row held by this lane
    const int col  = tile_n * 16 + lm;   // B/C/D column held by this lane

    const float scale = inv[arow];
    const float* arow_p = aggsum + (size_t)arow * DH;
    const float* xrow_p = xin    + (size_t)arow * DH;

    v8f acc = {};

    // (agg * inv) @ Wl , K = 128 in steps of 4
#pragma unroll
    for (int k = 0; k < DH; k += 4) {
        const int kk = k + 2 * hi;
        v2f a = *(const v2f*)(arow_p + kk);
        a.x *= scale; a.y *= scale;
        v2f b;
        b.x = Wl[(size_t)(kk    ) * DH + col];
        b.y = Wl[(size_t)(kk + 1) * DH + col];
        acc = __builtin_amdgcn_wmma_f32_16x16x4_f32(false, a, false, b,
                                                    (short)0, acc, false, false);
    }
    // x @ Wr
#pragma unroll
    for (int k = 0; k < DH; k += 4) {
        const int kk = k + 2 * hi;
        v2f a = *(const v2f*)(xrow_p + kk);
        v2f b;
        b.x = Wr[(size_t)(kk    ) * DH + col];
        b.y = Wr[(size_t)(kk + 1) * DH + col];
        acc = __builtin_amdgcn_wmma_f32_16x16x4_f32(false, a, false, b,
                                                    (short)0, acc, false, false);
    }

    const float bv = bias[col];
#pragma unroll
    for (int i = 0; i < 8; ++i) {
        float v = acc[i] + bv;
        v = v > 0.0f ? v : 0.0f;                       // ReLU
        out[(size_t)(row_base + hi * 8 + i) * DH + col] = v;
    }
}

// --------------------------------------------------------- final layer (DOUT = 1) + sigmoid
// One wave per node: dual dot-product over 128 features, shuffle reduce.
__global__ __launch_bounds__(256) void sage_final_kernel(const float* __restrict__ aggsum,
                                                         const float* __restrict__ xin,
                                                         const float* __restrict__ Wl,
                                                         const float* __restrict__ bias,
                                                         const float* __restrict__ Wr,
                                                         const float* __restrict__ inv,
                                                         float* __restrict__ out,
                                                         int nN) {
    const int lane = threadIdx.x & 31;
    const int n = blockIdx.x * 8 + (threadIdx.x >> 5);
    if (n >= nN) return;
    const float sc = inv[n];
    const float4 a  = *(const float4*)(aggsum + (size_t)n * DH + lane * 4);
    const float4 x  = *(const float4*)(xin    + (size_t)n * DH + lane * 4);
    const float4 wl = *(const float4*)(Wl + lane * 4);
    const float4 wr = *(const float4*)(Wr + lane * 4);
    float p = (a.x * sc) * wl.x + (a.y * sc) * wl.y +
              (a.z * sc) * wl.z + (a.w * sc) * wl.w +
              x.x * wr.x + x.y * wr.y + x.z * wr.z + x.w * wr.w;
#pragma unroll
    for (int off = 16; off > 0; off >>= 1) p += __shfl_xor(p, off, 32);
    if (lane == 0) {
        float v = p + bias[0];
        out[n] = 1.0f / (1.0f + expf(-v));
    }
}

// ----------------------------------------------------------------- launcher
extern "C" void kernel_launch(void* const* d_in, const int* in_sizes, int n_in,
                              void* d_out, int out_size, void* d_ws, size_t ws_size,
                              hipStream_t stream) {
    const float* x   = (const float*)d_in[0];
    const int*   ei  = (const int*)d_in[1];
    const float* Wl0 = (const float*)d_in[2];
    const float* b0  = (const float*)d_in[3];
    const float* Wr0 = (const float*)d_in[4];
    const float* Wl1 = (const float*)d_in[5];
    const float* b1  = (const float*)d_in[6];
    const float* Wr1 = (const float*)d_in[7];
    const float* Wl2 = (const float*)d_in[8];
    const float* b2  = (const float*)d_in[9];
    const float* Wr2 = (const float*)d_in[10];

    const int nN = in_sizes[0] / DIN;   // 100000
    const int nE = in_sizes[1] / 2;     // 3200000
    const int* src = ei;
    const int* dst = ei + nE;

    float* deg = (float*)d_ws;                      // nN floats (becomes inv-degree)
    float* agg = deg + (size_t)nN;                  // nN * 128
    float* h0  = agg + (size_t)nN * DH;             // nN * 128
    float* h1  = h0  + (size_t)nN * DH;             // nN * 128

    const dim3 blk(256);
    const dim3 gridE((nE + 7) / 8);                 // wave per edge
    const dim3 gridDeg((nE + 255) / 256);
    const dim3 gridN((nN + 255) / 256);
    const dim3 gridGemm((nN + 15) / 16);            // block = 16-row strip (8 n-tiles)
    const dim3 gridFinal((nN + 7) / 8);             // wave per node

    // inverse mean-degree (edges constant across layers -> compute once)
    hipMemsetAsync(deg, 0, (size_t)nN * sizeof(float), stream);
    deg_kernel<<<gridDeg, blk, 0, stream>>>(dst, deg, nE);
    inv_kernel<<<gridN, blk, 0, stream>>>(deg, nN);

    // ---- layer 0: x -> h0 (ReLU)
    hipMemsetAsync(agg, 0, (size_t)nN * DH * sizeof(float), stream);
    scatter_kernel<<<gridE, blk, 0, stream>>>(x, src, dst, agg, nE);
    sage_gemm_kernel<<<gridGemm, blk, 0, stream>>>(agg, x, Wl0, b0, Wr0, deg, h0, nN);

    // ---- layer 1: h0 -> h1 (ReLU)
    hipMemsetAsync(agg, 0, (size_t)nN * DH * sizeof(float), stream);
    scatter_kernel<<<gridE, blk, 0, stream>>>(h0, src, dst, agg, nE);
    sage_gemm_kernel<<<gridGemm, blk, 0, stream>>>(agg, h0, Wl1, b1, Wr1, deg, h1, nN);

    // ---- layer 2: h1 -> out (sigmoid, DOUT=1)
    hipMemsetAsync(agg, 0, (size_t)nN * DH * sizeof(float), stream);
    scatter_kernel<<<gridE, blk, 0, stream>>>(h1, src, dst, agg, nE);
    sage_final_kernel<<<gridFinal, blk, 0, stream>>>(agg, h1, Wl2, b2, Wr2, deg,
                                                     (float*)d_out, nN);
}